// GPURadiomics_30975304139415
// MI455X (gfx1250) — compile-verified
//
#include <hip/hip_runtime.h>
#include <math.h>

typedef __attribute__((ext_vector_type(2))) float v2f;
typedef __attribute__((ext_vector_type(8))) float v8f;

#define Bn   4
#define Dd   96
#define Hh   160
#define Ww   160
#define NVOX (Dd*Hh*Ww)          // 2,457,600 per batch item
#define NBINS 50

// conv tile geometry: 8 z x 8 y x 32 x voxels per block, halo'd 10x10x34 in LDS
#define TZ 8
#define TY 8
#define TX 32
#define LZ (TZ+2)
#define LY (TY+2)
#define LX (TX+2)
#define LDS_TILE (LZ*LY*LX)      // 3400 floats = 13.6 KB

// double accumulators per batch, stride 80:
// 0:n  1..4:S1..S4  5..12:gab[8]  13..18:lbp[6]
// 19..27: glcm [axis]{s,con,hom}  28..77: hist[50]
#define ACC_STRIDE 80
#define ACC_N     0
#define ACC_S1    1
#define ACC_GAB   5
#define ACC_LBP   13
#define ACC_GLCM  19
#define ACC_HIST  28

__global__ void init_kernel(double* acc) {
    int i = threadIdx.x;
    if (i < Bn * ACC_STRIDE) acc[i] = 0.0;
}

// ---------------- fused statistics pass: moments + histogram + glcm ----------
__global__ __launch_bounds__(256)
void stats_kernel(const float* __restrict__ x, const float* __restrict__ mask,
                  double* __restrict__ acc) {
    const int b = blockIdx.y;
    const float* xv = x    + (size_t)b * NVOX;
    const float* mv = mask + (size_t)b * NVOX;

    __shared__ double sacc[14];      // n,S1..S4, glcm 9
    __shared__ float  shist[NBINS];
    const int tid = threadIdx.x;
    for (int i = tid; i < 14;    i += blockDim.x) sacc[i]  = 0.0;
    for (int i = tid; i < NBINS; i += blockDim.x) shist[i] = 0.f;
    __syncthreads();

    double n = 0, S1 = 0, S2 = 0, S3 = 0, S4 = 0;
    double g[9];
    #pragma unroll
    for (int i = 0; i < 9; ++i) g[i] = 0;

    const int stride = gridDim.x * blockDim.x;
    for (int i = blockIdx.x * blockDim.x + tid; i < NVOX; i += stride) {
        __builtin_prefetch(&xv[i + stride], 0, 0);   // global_prefetch_b8
        __builtin_prefetch(&mv[i + stride], 0, 0);
        float v  = xv[i];
        float mf = mv[i];
        bool  m  = mf > 0.5f;
        float v2 = v * v;
        n  += mf;
        S1 += (double)(mf * v);
        S2 += (double)(mf * v2);
        S3 += (double)(mf * v2 * v);
        S4 += (double)(mf * v2 * v2);
        if (m && v >= -1.f && v <= 1.f) {
            int idx = (int)floorf((v + 1.f) * (NBINS * 0.5f));
            idx = idx < 0 ? 0 : (idx > NBINS - 1 ? NBINS - 1 : idx);
            atomicAdd(&shist[idx], mf);              // ds_add_f32
        }
        // glcm proxy: pair (p, p+1) along each axis
        int z  = i / (Hh * Ww);
        int r  = i - z * (Hh * Ww);
        int y  = r / Ww;
        int xc = r - y * Ww;
        if (m) {
            if (z < Dd - 1) {
                if (mv[i + Hh*Ww] > 0.5f) {
                    float d = v - xv[i + Hh*Ww];
                    g[0] += 1.0; g[1] += (double)(d*d); g[2] += (double)(1.f/(1.f+d*d));
                }
            }
            if (y < Hh - 1) {
                if (mv[i + Ww] > 0.5f) {
                    float d = v - xv[i + Ww];
                    g[3] += 1.0; g[4] += (double)(d*d); g[5] += (double)(1.f/(1.f+d*d));
                }
            }
            if (xc < Ww - 1) {
                if (mv[i + 1] > 0.5f) {
                    float d = v - xv[i + 1];
                    g[6] += 1.0; g[7] += (double)(d*d); g[8] += (double)(1.f/(1.f+d*d));
                }
            }
        }
    }

    atomicAdd(&sacc[0], n);
    atomicAdd(&sacc[1], S1); atomicAdd(&sacc[2], S2);
    atomicAdd(&sacc[3], S3); atomicAdd(&sacc[4], S4);
    #pragma unroll
    for (int i = 0; i < 9; ++i) atomicAdd(&sacc[5 + i], g[i]);
    __syncthreads();

    double* ab = acc + b * ACC_STRIDE;
    if (tid < 5)                 atomicAdd(&ab[ACC_N + tid], sacc[tid]);
    else if (tid < 14)           atomicAdd(&ab[ACC_GLCM + (tid - 5)], sacc[tid]);
    for (int i = tid; i < NBINS; i += blockDim.x)
        atomicAdd(&ab[ACC_HIST + i], (double)shist[i]);
}

// ---------------- WMMA conv: 8 gabor + 6 lbp filters in one 16x28 A matrix ---
__device__ __forceinline__ float loadA(const float* gf, const float* lf, int Mr, int k) {
    if (k >= 27) return 0.f;
    if (Mr < 8)  return gf[Mr * 27 + k];
    if (Mr < 14) return lf[(Mr - 8) * 27 + k];
    return 0.f;
}
__device__ __forceinline__ float loadB(const float* tile, int center, int k) {
    if (k >= 27) return 0.f;
    int dz = k / 9 - 1;
    int rm = k % 9;
    int dy = rm / 3 - 1;
    int dx = k % 3 - 1;
    return tile[center + (dz * LY + dy) * LX + dx];
}

__global__ __launch_bounds__(256)
void conv_wmma_kernel(const float* __restrict__ x, const float* __restrict__ mask,
                      const float* __restrict__ gf, const float* __restrict__ lf,
                      double* __restrict__ acc) {
    const int b = blockIdx.y;
    const int NTX = Ww / TX, NTY = Hh / TY;          // 5, 20 (NTZ = 12)
    int t  = blockIdx.x;
    int tx = t % NTX; t /= NTX;
    int ty = t % NTY;
    int tz = t / NTY;
    const int x0 = tx * TX, y0 = ty * TY, z0 = tz * TZ;

    const float* xv = x    + (size_t)b * NVOX;
    const float* mv = mask + (size_t)b * NVOX;

    __shared__ float tile[LDS_TILE];                 // halo'd 8x8x32 tile
    __shared__ float sgab[8];
    __shared__ float slbp[6];
    const int tid = threadIdx.x;
    if (tid < 8) sgab[tid] = 0.f;
    if (tid < 6) slbp[tid] = 0.f;

    for (int li = tid; li < LDS_TILE; li += 256) {
        int lz = li / (LY * LX);
        int rm = li % (LY * LX);
        int ly = rm / LX;
        int lx = rm % LX;
        int gz = z0 - 1 + lz, gy = y0 - 1 + ly, gx = x0 - 1 + lx;
        float val = 0.f;
        if ((unsigned)gz < (unsigned)Dd && (unsigned)gy < (unsigned)Hh &&
            (unsigned)gx < (unsigned)Ww)
            val = xv[(gz * Hh + gy) * Ww + gx];
        tile[li] = val;
    }
    __syncthreads();

    const int lane = tid & 31;
    const int wave = tid >> 5;
    const int M    = lane & 15;          // matrix row (filter id)
    const int h    = lane >> 4;          // half-wave selects K offset pair

    // A fragments: 16x28, rows 0..7 gabor, 8..13 lbp, 14..15 zero
    v2f afrag[7];
    #pragma unroll
    for (int s = 0; s < 7; ++s) {
        int k0 = 4 * s + 2 * h;
        afrag[s].x = loadA(gf, lf, M, k0);
        afrag[s].y = loadA(gf, lf, M, k0 + 1);
    }

    float facc[8];
    #pragma unroll
    for (int j = 0; j < 8; ++j) facc[j] = 0.f;

    // 128 wave-tiles of 16 voxels: (zr,yr) in 8x8 rows, xs in {0,1}
    for (int t2 = wave; t2 < (TZ * TY * TX) / 16; t2 += 8) {
        int zr = t2 >> 4;
        int yr = (t2 >> 1) & 7;
        int xs = t2 & 1;
        int xl = xs * 16 + M;                        // this lane's column voxel
        int center = ((zr + 1) * LY + (yr + 1)) * LX + (xl + 1);

        v8f c = {};
        #pragma unroll
        for (int s = 0; s < 7; ++s) {
            int k0 = 4 * s + 2 * h;
            v2f bfrag;
            bfrag.x = loadB(tile, center, k0);
            bfrag.y = loadB(tile, center, k0 + 1);
            c = __builtin_amdgcn_wmma_f32_16x16x4_f32(
                    false, afrag[s], false, bfrag, (short)0, c, false, false);
        }

        float mf = mv[((z0 + zr) * Hh + (y0 + yr)) * Ww + (x0 + xl)];
        if (h == 0) {                                // rows 0..7 = gabor
            #pragma unroll
            for (int j = 0; j < 8; ++j) facc[j] += fabsf(c[j]) * mf;
        } else {                                     // rows 8..13 = lbp
            #pragma unroll
            for (int j = 0; j < 6; ++j) facc[j] += (c[j] > 0.f) ? mf : 0.f;
        }
    }

    // reduce across the 16 lanes of each half-wave (xor masks < 16 keep halves apart)
    #pragma unroll
    for (int off = 8; off > 0; off >>= 1) {
        #pragma unroll
        for (int j = 0; j < 8; ++j)
            facc[j] += __shfl_xor(facc[j], off, 32);
    }
    if (lane == 0) {
        #pragma unroll
        for (int j = 0; j < 8; ++j) atomicAdd(&sgab[j], facc[j]);
    }
    if (lane == 16) {
        #pragma unroll
        for (int j = 0; j < 6; ++j) atomicAdd(&slbp[j], facc[j]);
    }
    __syncthreads();

    double* ab = acc + b * ACC_STRIDE;
    if (tid < 8)       atomicAdd(&ab[ACC_GAB + tid], (double)sgab[tid]);
    else if (tid < 14) atomicAdd(&ab[ACC_LBP + tid - 8], (double)slbp[tid - 8]);
}

// ---------------- finalize: 25 features per batch ---------------------------
__global__ void finalize_kernel(const double* __restrict__ acc, float* __restrict__ out) {
    int b = threadIdx.x;
    if (b >= Bn) return;
    const double* a = acc + b * ACC_STRIDE;
    double n  = a[ACC_N];
    double nn = fmax(n, 1.0);
    double S1 = a[1], S2 = a[2], S3 = a[3], S4 = a[4];
    double mu = S1 / nn;
    double d2 = S2 - 2.0*mu*S1 + mu*mu*n;
    double var = d2 / fmax(n - 1.0, 1.0);
    double sigma = sqrt(fmax(var, 0.0)) + 1e-8;
    double d3 = S3 - 3.0*mu*S2 + 3.0*mu*mu*S1 - mu*mu*mu*n;
    double d4 = S4 - 4.0*mu*S3 + 6.0*mu*mu*S2 - 4.0*mu*mu*mu*S1 + mu*mu*mu*mu*n;
    double skew = d3 / (sigma*sigma*sigma * nn);
    double kurt = d4 / (sigma*sigma*sigma*sigma * nn) - 3.0;

    double hsum = 0.0;
    for (int i = 0; i < NBINS; ++i) hsum += a[ACC_HIST + i];
    hsum += 1e-8;
    double ent = 0.0;
    for (int i = 0; i < NBINS; ++i) {
        double p = a[ACC_HIST + i] / hsum;
        ent -= p * log(p + 1e-8);
    }
    double valid = (n >= 10.0) ? 1.0 : 0.0;

    float* o = out + b * 25;
    o[0] = (float)(mu * valid);
    o[1] = (float)(sigma * valid);
    o[2] = (float)(skew * valid);
    o[3] = (float)(kurt * valid);
    o[4] = (float)(ent * valid);

    double lv = fmax(n, 1.0);
    for (int f = 0; f < 8; ++f) o[5 + f]  = (float)(a[ACC_GAB + f] / lv);
    for (int f = 0; f < 6; ++f) o[13 + f] = (float)(a[ACC_LBP + f] / lv);
    for (int ax = 0; ax < 3; ++ax) {
        double s  = a[ACC_GLCM + 3*ax + 0];
        double ss = fmax(s, 1.0);
        double ok = (s >= 4.0) ? 1.0 : 0.0;
        o[19 + 2*ax + 0] = (float)(a[ACC_GLCM + 3*ax + 1] / ss * ok);
        o[19 + 2*ax + 1] = (float)(a[ACC_GLCM + 3*ax + 2] / ss * ok);
    }
}

extern "C" void kernel_launch(void* const* d_in, const int* in_sizes, int n_in,
                              void* d_out, int out_size, void* d_ws, size_t ws_size,
                              hipStream_t stream) {
    const float* x    = (const float*)d_in[0];
    const float* mask = (const float*)d_in[1];
    const float* gf   = (const float*)d_in[2];   // 8*27
    const float* lf   = (const float*)d_in[3];   // 6*27
    float*  out = (float*)d_out;
    double* acc = (double*)d_ws;                 // 4*80 doubles = 2560 B

    init_kernel<<<1, Bn * ACC_STRIDE, 0, stream>>>(acc);
    stats_kernel<<<dim3(1200, Bn), 256, 0, stream>>>(x, mask, acc);
    conv_wmma_kernel<<<dim3(Bn * 0 + (Ww/TX)*(Hh/TY)*(Dd/TZ), Bn), 256, 0, stream>>>(x, mask, gf, lf, acc);
    finalize_kernel<<<1, 32, 0, stream>>>(acc, out);
}